// PGAHead_28690381538129
// MI455X (gfx1250) — compile-verified
//
#include <hip/hip_runtime.h>
#include <math.h>

typedef __attribute__((ext_vector_type(2))) float v2f;
typedef __attribute__((ext_vector_type(8))) float v8f;

#define BD 2048   // batch
#define DD 512    // feature dim
#define PD 768    // projection dim
#define NL 3      // layers

// ---------------------------------------------------------------------------
// Tiled WMMA f32 GEMM: C[M,N] = A[M,K] * B' (+ epilogue)
//   b_nn == 0 : B stored row-major [N,K]  -> C[m,n] = sum_k A[m,k]*B[n,k]
//   b_nn == 1 : B stored row-major [K,N]  -> C[m,n] = sum_k A[m,k]*B[k,n]
//   mode 0: plain store, 1: clip to [lo,hi], 2: add residual R[m,n]
// Block = 256 threads = 8 waves; tile M=128 x N=64; wave owns 32x32 output
// (4 independent 16x16 f32 accumulators -> 16 v_wmma per K-tile).
// Global->LDS staging is software-pipelined through registers: the next
// K-tile's global loads are issued before the WMMA block and drain during
// compute; they are only waited on at the following LDS commit.
// ---------------------------------------------------------------------------
__global__ __launch_bounds__(256) void gemm_wmma_f32(
    const float* __restrict__ A, const float* __restrict__ Bm,
    float* __restrict__ C, const float* __restrict__ R,
    int M, int N, int K, int b_nn, int mode, float clip_lo, float clip_hi)
{
  __shared__ float As[128][18];  // stride 18 floats (72B): rows stay 8B-aligned
  __shared__ float Bs[64][18];

  const int m0   = blockIdx.y * 128;
  const int n0   = blockIdx.x * 64;
  const int t    = threadIdx.x;
  const int wave = t >> 5;
  const int lane = t & 31;
  const int wm   = wave >> 1;        // 0..3 -> 32-row strip
  const int wn   = wave & 1;         // 0..1 -> 32-col strip
  const int fi   = lane & 15;        // M (A frag) / N (B frag) index
  const int kb   = (lane >> 4) * 2;  // K sub-base within fragment

  v8f acc00 = {}, acc01 = {}, acc10 = {}, acc11 = {};
  float ra[8], rb[4];

  // ---- prologue: stage tile kt=0 into registers (clause of loads) ----
#pragma unroll
  for (int i = 0; i < 8; ++i) {
    int e = t + i * 256;
    ra[i] = A[(size_t)(m0 + (e >> 4)) * K + (e & 15)];
  }
  if (b_nn) {
#pragma unroll
    for (int i = 0; i < 4; ++i) {
      int e = t + i * 256;
      rb[i] = Bm[(size_t)(e >> 6) * N + (n0 + (e & 63))];
    }
  } else {
#pragma unroll
    for (int i = 0; i < 4; ++i) {
      int e = t + i * 256;
      rb[i] = Bm[(size_t)(n0 + (e >> 4)) * K + (e & 15)];
    }
  }

  for (int kt = 0; kt < K; kt += 16) {
    // ---- commit staged registers to LDS ----
#pragma unroll
    for (int i = 0; i < 8; ++i) {
      int e = t + i * 256;
      As[e >> 4][e & 15] = ra[i];
    }
    if (b_nn) {
#pragma unroll
      for (int i = 0; i < 4; ++i) {
        int e = t + i * 256;
        Bs[e & 63][e >> 6] = rb[i];
      }
    } else {
#pragma unroll
      for (int i = 0; i < 4; ++i) {
        int e = t + i * 256;
        Bs[e >> 4][e & 15] = rb[i];
      }
    }
    __syncthreads();

    // ---- issue next K-tile's global loads (overlap with WMMA below) ----
    const int kn = kt + 16;
    if (kn < K) {
#pragma unroll
      for (int i = 0; i < 8; ++i) {
        int e = t + i * 256;
        ra[i] = A[(size_t)(m0 + (e >> 4)) * K + (kn + (e & 15))];
      }
      if (b_nn) {
#pragma unroll
        for (int i = 0; i < 4; ++i) {
          int e = t + i * 256;
          rb[i] = Bm[(size_t)(kn + (e >> 6)) * N + (n0 + (e & 63))];
        }
      } else {
#pragma unroll
        for (int i = 0; i < 4; ++i) {
          int e = t + i * 256;
          rb[i] = Bm[(size_t)(n0 + (e >> 4)) * K + (kn + (e & 15))];
        }
      }
    }
    // ---- near-cache prefetch of the tile after that ----
    if (kt + 32 < K) {
#pragma unroll
      for (int i = 0; i < 8; i += 4) {   // 2 lines/thread is plenty
        int e = t + i * 256;
        __builtin_prefetch(&A[(size_t)(m0 + (e >> 4)) * K + (kt + 32 + (e & 15))], 0, 3);
      }
    }

    // ---- 4 K-steps x 4 accumulators of v_wmma_f32_16x16x4_f32 ----
#pragma unroll
    for (int kk = 0; kk < 16; kk += 4) {
      v2f a0, a1, b0, b1;
      a0.x = As[wm * 32 + fi][kk + kb];
      a0.y = As[wm * 32 + fi][kk + kb + 1];
      a1.x = As[wm * 32 + 16 + fi][kk + kb];
      a1.y = As[wm * 32 + 16 + fi][kk + kb + 1];
      b0.x = Bs[wn * 32 + fi][kk + kb];
      b0.y = Bs[wn * 32 + fi][kk + kb + 1];
      b1.x = Bs[wn * 32 + 16 + fi][kk + kb];
      b1.y = Bs[wn * 32 + 16 + fi][kk + kb + 1];
      acc00 = __builtin_amdgcn_wmma_f32_16x16x4_f32(false, a0, false, b0,
                                                    (short)0, acc00, false, false);
      acc01 = __builtin_amdgcn_wmma_f32_16x16x4_f32(false, a0, false, b1,
                                                    (short)0, acc01, false, false);
      acc10 = __builtin_amdgcn_wmma_f32_16x16x4_f32(false, a1, false, b0,
                                                    (short)0, acc10, false, false);
      acc11 = __builtin_amdgcn_wmma_f32_16x16x4_f32(false, a1, false, b1,
                                                    (short)0, acc11, false, false);
    }
    __syncthreads();
  }

  // ---- epilogue: C/D layout: VGPR r -> M = r + 8*(lane>>4), N = lane&15 ----
  const int mb0 = m0 + wm * 32 + ((lane >> 4) << 3);   // M-subtile 0
  const int mb1 = mb0 + 16;                            // M-subtile 1
  const int nc0 = n0 + wn * 32 + fi;                   // N-subtile 0
#pragma unroll
  for (int r = 0; r < 8; ++r) {
    int ma = mb0 + r, mc = mb1 + r;
    float v00 = acc00[r], v01 = acc01[r], v10 = acc10[r], v11 = acc11[r];
    if (mode == 1) {
      v00 = fminf(fmaxf(v00, clip_lo), clip_hi);
      v01 = fminf(fmaxf(v01, clip_lo), clip_hi);
      v10 = fminf(fmaxf(v10, clip_lo), clip_hi);
      v11 = fminf(fmaxf(v11, clip_lo), clip_hi);
    } else if (mode == 2) {
      v00 += R[(size_t)ma * N + nc0];
      v01 += R[(size_t)ma * N + nc0 + 16];
      v10 += R[(size_t)mc * N + nc0];
      v11 += R[(size_t)mc * N + nc0 + 16];
    }
    C[(size_t)ma * N + nc0]      = v00;
    C[(size_t)ma * N + nc0 + 16] = v01;
    C[(size_t)mc * N + nc0]      = v10;
    C[(size_t)mc * N + nc0 + 16] = v11;
  }
}

// ---------------------------------------------------------------------------
// Row-wise L2 normalization: Y[row,:] = X[row,:] / max(||X[row,:]||, 1e-12)
// ---------------------------------------------------------------------------
__global__ __launch_bounds__(256) void l2norm_rows(
    const float* __restrict__ X, float* __restrict__ Y, int D)
{
  __shared__ float red[256];
  const int row = blockIdx.x;
  const float* x = X + (size_t)row * D;
  float s = 0.f;
  for (int j = threadIdx.x; j < D; j += 256) { float v = x[j]; s += v * v; }
  red[threadIdx.x] = s; __syncthreads();
  for (int o = 128; o > 0; o >>= 1) {
    if (threadIdx.x < o) red[threadIdx.x] += red[threadIdx.x + o];
    __syncthreads();
  }
  const float inv = 1.0f / fmaxf(sqrtf(red[0]), 1e-12f);
  float* y = Y + (size_t)row * D;
  for (int j = threadIdx.x; j < D; j += 256) y[j] = x[j] * inv;
}

// ---------------------------------------------------------------------------
// Top-8 same-label neighbors per row (matches jax top_k: value desc, index asc)
// ---------------------------------------------------------------------------
__global__ __launch_bounds__(256) void topk_rows(
    const float* __restrict__ S, const int* __restrict__ labels,
    int* __restrict__ idx_out, int B)
{
  __shared__ float sval[256];
  __shared__ int   sidx[256];
  const int row = blockIdx.x;
  const int lab = labels[row];
  float curV = 3.0e9f; int curI = -1;            // previously selected (none)
  for (int sel = 0; sel < 8; ++sel) {
    float bv = -4.0e9f; int bi = 0x7fffffff;
    for (int j = threadIdx.x; j < B; j += 256) {
      float v = (labels[j] == lab && j != row) ? S[(size_t)row * B + j] : -1.0e9f;
      bool elig = (v < curV) || (v == curV && j > curI);
      if (elig && (v > bv || (v == bv && j < bi))) { bv = v; bi = j; }
    }
    sval[threadIdx.x] = bv; sidx[threadIdx.x] = bi; __syncthreads();
    for (int o = 128; o > 0; o >>= 1) {
      if (threadIdx.x < o) {
        float v2 = sval[threadIdx.x + o]; int i2 = sidx[threadIdx.x + o];
        if (v2 > sval[threadIdx.x] ||
            (v2 == sval[threadIdx.x] && i2 < sidx[threadIdx.x])) {
          sval[threadIdx.x] = v2; sidx[threadIdx.x] = i2;
        }
      }
      __syncthreads();
    }
    curV = sval[0]; curI = sidx[0];
    if (threadIdx.x == 0) idx_out[row * 8 + sel] = curI;
    __syncthreads();
  }
}

// ---------------------------------------------------------------------------
// A = a * max(S*M, 0) + 1e-6*I  with M[i,j] = (j in knn(i)) | (i in knn(j))
// ---------------------------------------------------------------------------
__global__ __launch_bounds__(256) void build_A(
    const float* __restrict__ S, const int* __restrict__ idx,
    float* __restrict__ A, float a, int B)
{
  const size_t total = (size_t)B * B;
  for (size_t e = (size_t)blockIdx.x * 256 + threadIdx.x; e < total;
       e += (size_t)gridDim.x * 256) {
    int i = (int)(e / B), j = (int)(e % B);
    bool m = false;
#pragma unroll
    for (int t = 0; t < 8; ++t)
      m = m || (idx[i * 8 + t] == j) || (idx[j * 8 + t] == i);
    float v = m ? a * fmaxf(S[e], 0.f) : 0.f;
    if (i == j) v += 1e-6f;
    A[e] = v;
  }
}

__global__ __launch_bounds__(256) void rowsum_dinv(
    const float* __restrict__ A, float* __restrict__ dinv, int B)
{
  __shared__ float red[256];
  const int row = blockIdx.x;
  float s = 0.f;
  for (int j = threadIdx.x; j < B; j += 256) s += A[(size_t)row * B + j];
  red[threadIdx.x] = s; __syncthreads();
  for (int o = 128; o > 0; o >>= 1) {
    if (threadIdx.x < o) red[threadIdx.x] += red[threadIdx.x + o];
    __syncthreads();
  }
  if (threadIdx.x == 0) dinv[row] = rsqrtf(fmaxf(red[0], 1e-8f));
}

__global__ __launch_bounds__(256) void sym_scale(
    float* __restrict__ A, const float* __restrict__ dinv, int B)
{
  const size_t total = (size_t)B * B;
  for (size_t e = (size_t)blockIdx.x * 256 + threadIdx.x; e < total;
       e += (size_t)gridDim.x * 256) {
    int i = (int)(e / B), j = (int)(e % B);
    A[e] *= dinv[i] * dinv[j];
  }
}

// ---------------------------------------------------------------------------
// BatchNorm (biased var over rows) + ReLU, in place, one block per column
// ---------------------------------------------------------------------------
__global__ __launch_bounds__(256) void bn_relu(
    float* __restrict__ Z, const float* __restrict__ g,
    const float* __restrict__ b, int rows, int cols)
{
  __shared__ float rs[256], rq[256];
  const int col = blockIdx.x;
  float s = 0.f, q = 0.f;
  for (int r = threadIdx.x; r < rows; r += 256) {
    float v = Z[(size_t)r * cols + col];
    s += v; q += v * v;
  }
  rs[threadIdx.x] = s; rq[threadIdx.x] = q; __syncthreads();
  for (int o = 128; o > 0; o >>= 1) {
    if (threadIdx.x < o) {
      rs[threadIdx.x] += rs[threadIdx.x + o];
      rq[threadIdx.x] += rq[threadIdx.x + o];
    }
    __syncthreads();
  }
  const float inv_n = 1.0f / (float)rows;
  const float mu  = rs[0] * inv_n;
  const float var = rq[0] * inv_n - mu * mu;
  const float sc  = g[col] * rsqrtf(var + 1e-5f);
  const float sh  = b[col] - mu * sc;
  for (int r = threadIdx.x; r < rows; r += 256) {
    size_t e = (size_t)r * cols + col;
    Z[e] = fmaxf(Z[e] * sc + sh, 0.f);
  }
}

// ---------------------------------------------------------------------------
// Loss accumulators
// ---------------------------------------------------------------------------
__global__ __launch_bounds__(256) void mse_acc(
    const float* __restrict__ X, const float* __restrict__ Y,
    size_t N, float scale, float* __restrict__ acc)
{
  __shared__ float red[256];
  float s = 0.f;
  for (size_t e = (size_t)blockIdx.x * 256 + threadIdx.x; e < N;
       e += (size_t)gridDim.x * 256) {
    float d = X[e] - Y[e]; s += d * d;
  }
  red[threadIdx.x] = s; __syncthreads();
  for (int o = 128; o > 0; o >>= 1) {
    if (threadIdx.x < o) red[threadIdx.x] += red[threadIdx.x + o];
    __syncthreads();
  }
  if (threadIdx.x == 0) atomicAdd(acc, red[0] * scale);
}

__global__ __launch_bounds__(256) void idea_acc(
    const float* __restrict__ Kmat, const int* __restrict__ labels,
    int B, float scale, float* __restrict__ acc)
{
  __shared__ float red[256];
  const size_t total = (size_t)B * B;
  float s = 0.f;
  for (size_t e = (size_t)blockIdx.x * 256 + threadIdx.x; e < total;
       e += (size_t)gridDim.x * 256) {
    int i = (int)(e / B), j = (int)(e % B);
    float tgt = (labels[i] == labels[j]) ? 0.99f : 0.0f;
    float d = Kmat[e] - tgt; s += d * d;
  }
  red[threadIdx.x] = s; __syncthreads();
  for (int o = 128; o > 0; o >>= 1) {
    if (threadIdx.x < o) red[threadIdx.x] += red[threadIdx.x + o];
    __syncthreads();
  }
  if (threadIdx.x == 0) atomicAdd(acc, red[0] * scale);
}

__global__ void init_out(float* out) {
  out[0] = 0.f; out[1] = 0.f; out[2] = 0.f; out[3] = 0.f;
}
__global__ void finalize_out(float* out) {
  out[3] = 64.0f * out[0] + 16.0f * out[1] + out[2];
}

// ---------------------------------------------------------------------------
extern "C" void kernel_launch(void* const* d_in, const int* in_sizes, int n_in,
                              void* d_out, int out_size, void* d_ws, size_t ws_size,
                              hipStream_t stream)
{
  (void)in_sizes; (void)n_in; (void)out_size; (void)ws_size;
  const float* feats  = (const float*)d_in[0];  // [3,2048,512]
  const int*   labels = (const int*)  d_in[1];  // [2048]
  const float* W1     = (const float*)d_in[2];  // [3,512,512]
  const float* W2     = (const float*)d_in[3];  // [3,512,512]
  const float* gamma  = (const float*)d_in[4];  // [3,512]
  const float* beta   = (const float*)d_in[5];  // [3,512]
  const float* Wproj  = (const float*)d_in[6];  // [768,512]
  float* out = (float*)d_out;                   // [4] f32

  const size_t SZ_XN = (size_t)BD * DD;         // 2048*512
  const size_t SZ_S  = (size_t)BD * BD;         // 2048*2048
  const size_t SZ_P  = (size_t)BD * PD;         // 2048*768

  float* ws  = (float*)d_ws;
  float* Xn  = ws;                 // normalized feats
  float* S   = Xn + SZ_XN;         // cosine sim (clipped)
  float* K0  = S  + SZ_S;          // A / A_norm ping
  float* K1  = K0 + SZ_S;          // A / A_norm pong
  float* T   = K1 + SZ_S;          // X@W1^T / Zr@W2^T
  float* H   = T  + SZ_XN;         // A_norm@T (pre/post BN)
  float* Z   = H  + SZ_XN;         // layer output
  float* P0  = Z  + SZ_XN;         // projection ping
  float* P1  = P0 + SZ_P;          // projection pong
  float* dv  = P1 + SZ_P;          // d^-1/2 [2048]
  int*   idx = (int*)(dv + BD);    // top-8 indices [2048*8]

  const dim3 blk(256);
  const dim3 gS (BD / 64, BD / 128);   // C: 2048x2048
  const dim3 gD (DD / 64, BD / 128);   // C: 2048x512
  const dim3 gP (PD / 64, BD / 128);   // C: 2048x768
  const float clip_lo = -1.0f + 1e-8f, clip_hi = 1.0f - 1e-8f;

  init_out<<<1, 1, 0, stream>>>(out);

  for (int L = 0; L < NL; ++L) {
    const float* X  = feats + (size_t)L * SZ_XN;
    const float* w1 = W1 + (size_t)L * DD * DD;
    const float* w2 = W2 + (size_t)L * DD * DD;
    float* Kc = (L & 1) ? K1 : K0;
    float* Kp = (L & 1) ? K0 : K1;
    float* Pc = (L & 1) ? P1 : P0;
    float* Pp = (L & 1) ? P0 : P1;
    const float aL = 1.0f + 0.1f * (float)L;   // linspace(1.0, 1.2, 3)

    // --- graph: K = A_norm ---
    l2norm_rows<<<BD, blk, 0, stream>>>(X, Xn, DD);
    gemm_wmma_f32<<<gS, blk, 0, stream>>>(Xn, Xn, S, nullptr,
                                          BD, BD, DD, 0, 1, clip_lo, clip_hi);
    topk_rows<<<BD, blk, 0, stream>>>(S, labels, idx, BD);
    build_A<<<16384, blk, 0, stream>>>(S, idx, Kc, aL, BD);
    rowsum_dinv<<<BD, blk, 0, stream>>>(Kc, dv, BD);
    sym_scale<<<16384, blk, 0, stream>>>(Kc, dv, BD);

    // --- GAM: Z = A@relu(BN(A@(X@W1^T)))@W2^T + X ---
    gemm_wmma_f32<<<gD, blk, 0, stream>>>(X, w1, T, nullptr,
                                          BD, DD, DD, 0, 0, 0.f, 0.f);
    gemm_wmma_f32<<<gD, blk, 0, stream>>>(Kc, T, H, nullptr,
                                          BD, DD, BD, 1, 0, 0.f, 0.f);
    bn_relu<<<DD, blk, 0, stream>>>(H, gamma + L * DD, beta + L * DD, BD, DD);
    gemm_wmma_f32<<<gD, blk, 0, stream>>>(H, w2, T, nullptr,
                                          BD, DD, DD, 0, 0, 0.f, 0.f);
    gemm_wmma_f32<<<gD, blk, 0, stream>>>(Kc, T, Z, X,
                                          BD, DD, BD, 1, 2, 0.f, 0.f);

    // --- projection + row l2norm ---
    gemm_wmma_f32<<<gP, blk, 0, stream>>>(Z, Wproj, Pc, nullptr,
                                          BD, PD, DD, 0, 0, 0.f, 0.f);
    l2norm_rows<<<BD, blk, 0, stream>>>(Pc, Pc, PD);

    // --- streaming losses ---
    if (L > 0) {
      mse_acc<<<1024, blk, 0, stream>>>(Kp, Kc, SZ_S, 1.0f / (float)SZ_S, out + 0);
      mse_acc<<<1024, blk, 0, stream>>>(Pp, Pc, SZ_P, 1.0f / (float)SZ_P, out + 1);
    }
    if (L == NL - 1) {
      idea_acc<<<1024, blk, 0, stream>>>(Kc, labels, BD, 1.0f / (float)SZ_S, out + 2);
    }
  }

  finalize_out<<<1, 1, 0, stream>>>(out);
}